// CoSSL_58179626992261
// MI455X (gfx1250) — compile-verified
//
#include <hip/hip_runtime.h>
#include <hip/hip_bf16.h>

typedef float v2f __attribute__((ext_vector_type(2)));
typedef float v8f __attribute__((ext_vector_type(8)));

// ---------------- workspace layout (float offsets) ----------------
#define OFF_BUFA 0L                       // 16,777,216 floats (conv out, max layer)
#define OFF_BUFB 16777216L                // 16,777,216 floats (pool out / layer IO)
#define OFF_FEAT 33554432L                // 2*128*128 (g=0:k, g=1:q)
#define OFF_SPOS 33587200L                // 128
#define OFF_REFN 33587328L                // 128*2304
#define OFF_SNEG 33882240L                // 128*2048
#define OFF_SREF 34144384L                // 128*2048
#define OFF_MEAN 34406528L                // up to 128
#define OFF_RSTD 34406656L                // up to 128

// ---------------- direct conv (stride 2, pad 1, 3x3) + bias ----------------
__global__ void conv_kernel(const float* __restrict__ in, const float* __restrict__ w,
                            const float* __restrict__ bias, float* __restrict__ out,
                            int ci, int Hi, int Wi, int co, int Ho, int Wo, long inNStride)
{
    long idx = (long)blockIdx.x * blockDim.x + threadIdx.x;
    long total = 128L * co * Ho * Wo;
    if (idx >= total) return;
    int wo_ = (int)(idx % Wo);
    int ho_ = (int)((idx / Wo) % Ho);
    int c   = (int)((idx / ((long)Wo * Ho)) % co);
    int n   = (int)(idx / ((long)Wo * Ho * co));

    float acc = bias[c];
    const float* inN = in + (long)n * inNStride;
    for (int cc = 0; cc < ci; ++cc) {
        const float* inC = inN + (long)cc * Hi * Wi;
        const float* wc  = w + ((long)c * ci + cc) * 9;
        #pragma unroll
        for (int kh = 0; kh < 3; ++kh) {
            int hi = 2 * ho_ + kh - 1;
            if (hi < 0 || hi >= Hi) continue;
            #pragma unroll
            for (int kw = 0; kw < 3; ++kw) {
                int wi = 2 * wo_ + kw - 1;
                if (wi < 0 || wi >= Wi) continue;
                acc += inC[(long)hi * Wi + wi] * wc[kh * 3 + kw];
            }
        }
    }
    out[idx] = acc;
}

// ---------------- per-channel batch-norm statistics over (N,H,W) ----------------
__global__ void stats_kernel(const float* __restrict__ y, float* __restrict__ mean,
                             float* __restrict__ rstd, int co, int HW)
{
    int c = blockIdx.x;
    long total = 128L * HW;
    float s = 0.f, ss = 0.f;
    for (long i = threadIdx.x; i < total; i += blockDim.x) {
        long n = i / HW, r = i % HW;
        float v = y[(n * co + c) * (long)HW + r];
        s += v; ss += v * v;
    }
    __shared__ float sh1[256], sh2[256];
    sh1[threadIdx.x] = s; sh2[threadIdx.x] = ss;
    __syncthreads();
    for (int off = blockDim.x / 2; off > 0; off >>= 1) {
        if ((int)threadIdx.x < off) {
            sh1[threadIdx.x] += sh1[threadIdx.x + off];
            sh2[threadIdx.x] += sh2[threadIdx.x + off];
        }
        __syncthreads();
    }
    if (threadIdx.x == 0) {
        float m = sh1[0] / (float)total;
        float v = sh2[0] / (float)total - m * m;
        mean[c] = m;
        rstd[c] = rsqrtf(v + 1e-5f);
    }
}

// ---------------- fused BN-normalize + ReLU + 3x3 avgpool (stride 1, pad 1) ----------------
__global__ void normpool_kernel(const float* __restrict__ y, const float* __restrict__ mean,
                                const float* __restrict__ rstd, const float* __restrict__ gam,
                                const float* __restrict__ bet, float* __restrict__ out,
                                int co, int Ho, int Wo)
{
    long idx = (long)blockIdx.x * blockDim.x + threadIdx.x;
    long total = 128L * co * Ho * Wo;
    if (idx >= total) return;
    int wo_ = (int)(idx % Wo);
    int ho_ = (int)((idx / Wo) % Ho);
    int c   = (int)((idx / ((long)Wo * Ho)) % co);
    int n   = (int)(idx / ((long)Wo * Ho * co));

    float m = mean[c], rs = rstd[c], g = gam[c], b = bet[c];
    const float* yc = y + ((long)n * co + c) * (long)Ho * Wo;
    float s = 0.f;
    #pragma unroll
    for (int dh = -1; dh <= 1; ++dh) {
        int h2 = ho_ + dh;
        if (h2 < 0 || h2 >= Ho) continue;
        #pragma unroll
        for (int dw = -1; dw <= 1; ++dw) {
            int w2 = wo_ + dw;
            if (w2 < 0 || w2 >= Wo) continue;
            float v = (yc[(long)h2 * Wo + w2] - m) * rs * g + b;
            s += fmaxf(v, 0.f);
        }
    }
    out[idx] = s * (1.f / 9.f);
}

// ---------------- global spatial mean: (128,128,64,8) -> (128,128) ----------------
__global__ void gmean_kernel(const float* __restrict__ in, float* __restrict__ feat, int HW)
{
    int idx = blockIdx.x * blockDim.x + threadIdx.x; // 128*128 threads
    if (idx >= 128 * 128) return;
    const float* p = in + (long)idx * HW;
    float s = 0.f;
    for (int i = 0; i < HW; ++i) s += p[i];
    feat[idx] = s / (float)HW;
}

// ---------------- L2-normalize q,k rows + score_pos dot ----------------
__global__ void l2pos_kernel(float* __restrict__ feat, float* __restrict__ spos)
{
    int n = blockIdx.x, c = threadIdx.x; // 128 threads
    float k = feat[n * 128 + c];
    float q = feat[128 * 128 + n * 128 + c];
    __shared__ float sh[128];

    sh[c] = k * k; __syncthreads();
    for (int off = 64; off > 0; off >>= 1) { if (c < off) sh[c] += sh[c + off]; __syncthreads(); }
    float nk = fmaxf(sqrtf(sh[0]), 1e-12f); __syncthreads();

    sh[c] = q * q; __syncthreads();
    for (int off = 64; off > 0; off >>= 1) { if (c < off) sh[c] += sh[c + off]; __syncthreads(); }
    float nq = fmaxf(sqrtf(sh[0]), 1e-12f); __syncthreads();

    k /= nk; q /= nq;
    feat[n * 128 + c] = k;
    feat[128 * 128 + n * 128 + c] = q;

    sh[c] = q * k; __syncthreads();
    for (int off = 64; off > 0; off >>= 1) { if (c < off) sh[c] += sh[c + off]; __syncthreads(); }
    if (c == 0) spos[n] = sh[0];
}

// ---------------- L2-normalize ref rows (2304 wide) ----------------
__global__ void refnorm_kernel(const float* __restrict__ rf, float* __restrict__ rn)
{
    int n = blockIdx.x;
    const float* row = rf + (long)n * 2304;
    float s = 0.f;
    for (int i = threadIdx.x; i < 2304; i += blockDim.x) { float v = row[i]; s += v * v; }
    __shared__ float sh[256];
    sh[threadIdx.x] = s; __syncthreads();
    for (int off = 128; off > 0; off >>= 1) {
        if ((int)threadIdx.x < off) sh[threadIdx.x] += sh[threadIdx.x + off];
        __syncthreads();
    }
    float nrm = fmaxf(sqrtf(sh[0]), 1e-12f);
    for (int i = threadIdx.x; i < 2304; i += blockDim.x) rn[(long)n * 2304 + i] = row[i] / nrm;
}

// ---------------- fp32 WMMA GEMM: C[MxN] = A[MxK] * B[KxN] (row-major) ----------------
// One wave per 16x32 strip: A fragment loaded once (b64) and reused for two
// V_WMMA_F32_16X16X4_F32 per K-step; B rows prefetched 16 rows (four K-steps) ahead.
__global__ void gemm_wmma_f32(const float* __restrict__ A, const float* __restrict__ B,
                              float* __restrict__ C, int M, int N, int K,
                              int lda, int ldb, int ldc)
{
    int wave = threadIdx.x >> 5;
    int lane = threadIdx.x & 31;
    int half = lane >> 4;      // 0: lanes 0-15, 1: lanes 16-31
    int lr   = lane & 15;
    int n0 = (blockIdx.x * (blockDim.x >> 5) + wave) * 32;   // 32-wide strip
    int m0 = blockIdx.y * 16;
    if (n0 >= N || m0 >= M) return;     // wave-uniform

    const float* Arow  = A + (long)(m0 + lr) * lda;   // lanes 0-31 cover M=lr, halves split K
    const float* Bcol0 = B + n0 + lr;                 // lanes cover N=lr, halves split K
    const float* Bcol1 = B + n0 + 16 + lr;

    v8f acc0 = {0.f, 0.f, 0.f, 0.f, 0.f, 0.f, 0.f, 0.f};
    v8f acc1 = {0.f, 0.f, 0.f, 0.f, 0.f, 0.f, 0.f, 0.f};
    for (int k = 0; k < K; k += 4) {
        // contiguous K,K+1 pair -> single 64-bit load
        v2f a = *(const v2f*)(Arow + k + 2 * half);          // VGPR0: K=0/K=2, VGPR1: K=1/K=3
        long r0 = (long)(k + 2 * half + 0) * ldb;
        long r1 = (long)(k + 2 * half + 1) * ldb;
        v2f b0, b1;
        b0.x = Bcol0[r0]; b0.y = Bcol0[r1];
        b1.x = Bcol1[r0]; b1.y = Bcol1[r1];
        // speculative prefetch of the B rows 16 rows (four K-steps) ahead (global_prefetch_b8)
        __builtin_prefetch(Bcol0 + r0 + 16L * ldb, 0, 1);
        acc0 = __builtin_amdgcn_wmma_f32_16x16x4_f32(false, a, false, b0,
                                                     (short)0, acc0, false, false);
        acc1 = __builtin_amdgcn_wmma_f32_16x16x4_f32(false, a, false, b1,
                                                     (short)0, acc1, false, false);
    }
    // C/D: VGPR i -> row m0 + i + 8*half, col n0 + lr
    float* Crow0 = C + (long)(m0 + 8 * half) * ldc + n0 + lr;
    float* Crow1 = Crow0 + 16;
    #pragma unroll
    for (int i = 0; i < 8; ++i) {
        Crow0[(long)i * ldc] = acc0[i];
        Crow1[(long)i * ldc] = acc1[i];
    }
}

// ---------------- top-5 (masked) + final score/mask assembly, one block per row ----------------
__global__ void finalize_kernel(const float* __restrict__ sneg, const float* __restrict__ sref,
                                const float* __restrict__ spos, const int* __restrict__ indices,
                                const float* __restrict__ idxq, float* __restrict__ outScore,
                                float* __restrict__ outMask)
{
    int r = blockIdx.x;
    int tid = threadIdx.x; // 256 threads
    float myIdx = (float)indices[r];
    const float* srow = sref + (long)r * 2048;

    __shared__ float sv[256];
    __shared__ int   si[256];
    __shared__ int   chosen[5];

    for (int t = 0; t < 5; ++t) {
        float bv = -__builtin_inff();
        int bi = 1 << 30;
        for (int c = tid; c < 2048; c += 256) {
            bool skip = false;
            for (int j = 0; j < t; ++j) if (chosen[j] == c) skip = true;
            if (skip) continue;
            float v = (idxq[c] == myIdx) ? -__builtin_inff() : srow[c];
            if (v > bv || (v == bv && c < bi)) { bv = v; bi = c; }
        }
        sv[tid] = bv; si[tid] = bi;
        __syncthreads();
        for (int off = 128; off > 0; off >>= 1) {
            if (tid < off) {
                float ov = sv[tid + off]; int oi = si[tid + off];
                if (ov > sv[tid] || (ov == sv[tid] && oi < si[tid])) { sv[tid] = ov; si[tid] = oi; }
            }
            __syncthreads();
        }
        if (tid == 0) chosen[t] = si[0];
        __syncthreads();
    }

    if (tid == 0) {
        outScore[(long)r * 2049] = spos[r];
        outMask[(long)r * 2049]  = 1.f;
    }
    for (int c = tid; c < 2048; c += 256) {
        bool ms = (idxq[c] == myIdx);
        bool top = false;
        #pragma unroll
        for (int j = 0; j < 5; ++j) if (chosen[j] == c) top = true;
        float sr2 = ms ? 1.f : srow[c];
        float wgt = top ? 1.f : -1.f;
        outScore[(long)r * 2049 + 1 + c] = sneg[(long)r * 2048 + c] * sr2 * wgt;
        outMask[(long)r * 2049 + 1 + c]  = top ? 1.f : (ms ? 1.f : 0.f);
    }
}

// ---------------- host driver ----------------
extern "C" void kernel_launch(void* const* d_in, const int* in_sizes, int n_in,
                              void* d_out, int out_size, void* d_ws, size_t ws_size,
                              hipStream_t stream)
{
    const float* feats     = (const float*)d_in[0];
    const float* ref_feats = (const float*)d_in[1];
    const int*   indices   = (const int*)d_in[2];
    const float* moco      = (const float*)d_in[3];
    const float* refq      = (const float*)d_in[4];
    const float* idxq      = (const float*)d_in[5];
    const float* W[4]  = { (const float*)d_in[6],  (const float*)d_in[10], (const float*)d_in[14], (const float*)d_in[18] };
    const float* Bp[4] = { (const float*)d_in[7],  (const float*)d_in[11], (const float*)d_in[15], (const float*)d_in[19] };
    const float* G[4]  = { (const float*)d_in[8],  (const float*)d_in[12], (const float*)d_in[16], (const float*)d_in[20] };
    const float* Be[4] = { (const float*)d_in[9],  (const float*)d_in[13], (const float*)d_in[17], (const float*)d_in[21] };

    float* ws   = (float*)d_ws;
    float* bufA = ws + OFF_BUFA;
    float* bufB = ws + OFF_BUFB;
    float* feat = ws + OFF_FEAT;
    float* spos = ws + OFF_SPOS;
    float* refn = ws + OFF_REFN;
    float* sneg = ws + OFF_SNEG;
    float* sref = ws + OFF_SREF;
    float* mean = ws + OFF_MEAN;
    float* rstd = ws + OFF_RSTD;

    static const int ciA[4] = { 1, 4, 16, 64 };
    static const int coA[4] = { 4, 16, 64, 128 };
    static const int HiA[4] = { 1024, 512, 256, 128 };
    static const int WiA[4] = { 128, 64, 32, 16 };

    for (int g = 0; g < 2; ++g) {               // g=0: k-encoder (feats[:,0]), g=1: q-encoder (feats[:,1])
        const float* inPtr = feats + (long)g * 1024 * 128;
        long inNStride = 2L * 1024 * 128;
        for (int l = 0; l < 4; ++l) {
            int Ho = HiA[l] / 2, Wo = WiA[l] / 2;
            long total = 128L * coA[l] * Ho * Wo;
            int blocks = (int)((total + 255) / 256);
            conv_kernel<<<blocks, 256, 0, stream>>>(inPtr, W[l], Bp[l], bufA,
                                                    ciA[l], HiA[l], WiA[l], coA[l], Ho, Wo, inNStride);
            stats_kernel<<<coA[l], 256, 0, stream>>>(bufA, mean, rstd, coA[l], Ho * Wo);
            normpool_kernel<<<blocks, 256, 0, stream>>>(bufA, mean, rstd, G[l], Be[l], bufB,
                                                        coA[l], Ho, Wo);
            inPtr = bufB;
            inNStride = (long)coA[l] * Ho * Wo;
        }
        gmean_kernel<<<64, 256, 0, stream>>>(bufB, feat + (long)g * 128 * 128, 512);
    }

    l2pos_kernel<<<128, 128, 0, stream>>>(feat, spos);
    refnorm_kernel<<<128, 256, 0, stream>>>(ref_feats, refn);

    // score_neg = q @ MoCoQueue   (q = feat[g=1], 128x128; B = 128x2048)
    // 8 waves/block, 32 cols per wave -> block covers 256 cols
    gemm_wmma_f32<<<dim3(8, 8), 256, 0, stream>>>(feat + 128 * 128, moco, sneg,
                                                  128, 2048, 128, 128, 2048, 2048);
    // score_ref = ref_n @ RefQueue (128x2304 by 2304x2048)
    gemm_wmma_f32<<<dim3(8, 8), 256, 0, stream>>>(refn, refq, sref,
                                                  128, 2048, 2304, 2304, 2048, 2048);

    float* outScore = (float*)d_out;
    float* outMask  = (float*)d_out + 128L * 2049;
    finalize_kernel<<<128, 256, 0, stream>>>(sneg, sref, spos, indices, idxq, outScore, outMask);
}